// ModelNew_29497835389779
// MI455X (gfx1250) — compile-verified
//
#include <hip/hip_runtime.h>

typedef __attribute__((ext_vector_type(2))) float v2f;
typedef __attribute__((ext_vector_type(8))) float v8f;

#define I_DIM   8192
#define B_DIM   8192
#define H_DIM   8192
#define OUT_SCALE 0.75f   // SCALING_FACTOR * 0.5

// ---------------------------------------------------------------------------
// Phase 0: zero the colsum accumulator in workspace (harness poisons d_ws).
// ---------------------------------------------------------------------------
__global__ void zero_colsum_kernel(float* __restrict__ colsum) {
    int i = blockIdx.x * blockDim.x + threadIdx.x;
    if (i < I_DIM) colsum[i] = 0.0f;
}

// ---------------------------------------------------------------------------
// Phase 1: colsum[i] = sum_h W[h][i].
// Grid: (8, 128). Each block: 256 threads * float4 = 1024 columns, 64 rows.
// float4 loads are fully coalesced (consecutive threads -> consecutive 16B).
// Partial stripe sums combined with global f32 atomics (128 adds/address).
// ---------------------------------------------------------------------------
__global__ __launch_bounds__(256) void colsum_kernel(const float* __restrict__ W,
                                                     float* __restrict__ colsum,
                                                     int rows_per_stripe) {
    const int col4 = blockIdx.x * 256 + threadIdx.x;        // float4 column index
    const int row0 = blockIdx.y * rows_per_stripe;
    const int ld4  = I_DIM / 4;

    const float4* Wv = (const float4*)W;
    size_t base = (size_t)row0 * ld4 + col4;

    float4 acc = make_float4(0.f, 0.f, 0.f, 0.f);
    #pragma unroll 8
    for (int r = 0; r < rows_per_stripe; ++r) {
        float4 w = Wv[base + (size_t)r * ld4];
        acc.x += w.x; acc.y += w.y; acc.z += w.z; acc.w += w.w;
    }

    float* dst = colsum + (size_t)col4 * 4;
    atomicAdd(dst + 0, acc.x);
    atomicAdd(dst + 1, acc.y);
    atomicAdd(dst + 2, acc.z);
    atomicAdd(dst + 3, acc.w);
}

// ---------------------------------------------------------------------------
// Phase 2: out[b] = (x[b,:] . colsum) * 0.75 using V_WMMA_F32_16X16X4_F32.
//
// One block per 16-row group; 8 waves split the K=8192 dimension (1024 each).
// A tile (16x4 f32, 2 VGPRs/lane):
//   lanes 0-15 : row M=lane, K = k+0, k+1
//   lanes 16-31: row M=lane-16, K = k+2, k+3
// B tile (4x16): every column = colsum[k..k+3], so every column of D carries
// the same 16 running dot products. Column redundancy is free: the kernel is
// HBM-bound, the matrix pipe has slack.
// D layout: VGPR r, lanes 0-15 -> M=r; lanes 16-31 -> M=8+r (any N).
// Wave partials merged through LDS ds_add_f32 atomics.
// ---------------------------------------------------------------------------
__global__ __launch_bounds__(256) void matvec_wmma_kernel(const float* __restrict__ X,
                                                          const float* __restrict__ v,
                                                          float* __restrict__ out) {
    __shared__ float acc_lds[16];

    const int tid  = threadIdx.x;
    const int lane = tid & 31;
    const int wave = tid >> 5;                 // 0..7

    if (tid < 16) acc_lds[tid] = 0.0f;
    __syncthreads();

    const int rowbase = blockIdx.x * 16;
    const int kchunk  = I_DIM / 8;             // 1024 K per wave
    const int k0      = wave * kchunk;

    const int m     = lane & 15;               // row within the 16-row group
    const int khalf = (lane >> 4) * 2;         // 0 for lanes 0-15, 2 for 16-31

    const float* aptr = X + (size_t)(rowbase + m) * I_DIM + k0 + khalf;
    const float* bptr = v + k0 + khalf;

    v8f c0 = {};
    v8f c1 = {};
    // 256 WMMAs per wave, two interleaved accumulators to pipeline D->C.
    for (int k = 0; k < kchunk; k += 8) {
        v2f a0 = *(const v2f*)(aptr + k);
        v2f b0 = *(const v2f*)(bptr + k);
        v2f a1 = *(const v2f*)(aptr + k + 4);
        v2f b1 = *(const v2f*)(bptr + k + 4);
        c0 = __builtin_amdgcn_wmma_f32_16x16x4_f32(
                false, a0, false, b0, (short)0, c0, false, false);
        c1 = __builtin_amdgcn_wmma_f32_16x16x4_f32(
                false, a1, false, b1, (short)0, c1, false, false);
    }

    // lane 0 holds rows 0..7 in c[0..7]; lane 16 holds rows 8..15.
    if (m == 0) {
        const int mbase = (lane >> 4) * 8;
        #pragma unroll
        for (int r = 0; r < 8; ++r) {
            atomicAdd(&acc_lds[mbase + r], c0[r] + c1[r]);
        }
    }
    __syncthreads();

    if (tid < 16) out[rowbase + tid] = acc_lds[tid] * OUT_SCALE;
}

// ---------------------------------------------------------------------------
// Launcher: zero scratch -> column sum -> WMMA matvec, all on `stream`.
// ---------------------------------------------------------------------------
extern "C" void kernel_launch(void* const* d_in, const int* in_sizes, int n_in,
                              void* d_out, int out_size, void* d_ws, size_t ws_size,
                              hipStream_t stream) {
    const float* x = (const float*)d_in[0];   // (8192, 8192) f32
    const float* w = (const float*)d_in[1];   // (8192, 8192) f32
    float* out     = (float*)d_out;           // (8192, 1) f32
    float* colsum  = (float*)d_ws;            // 8192 f32 = 32 KB scratch

    zero_colsum_kernel<<<(I_DIM + 255) / 256, 256, 0, stream>>>(colsum);

    dim3 g1(I_DIM / 1024, 128);               // (8, 128); 64 rows per stripe
    colsum_kernel<<<g1, 256, 0, stream>>>(w, colsum, H_DIM / 128);

    matvec_wmma_kernel<<<B_DIM / 16, 256, 0, stream>>>(x, colsum, out);
}